// NodeHyperedgeAttention_17712445129261
// MI455X (gfx1250) — compile-verified
//
#include <hip/hip_runtime.h>

#define DIM 128
#define NH 4
#define HD 32

typedef __attribute__((ext_vector_type(2))) float v2f;
typedef __attribute__((ext_vector_type(8))) float v8f;

// Monotone order-preserving float<->uint encoding for atomicMax-based segment max.
__device__ __forceinline__ unsigned fenc(float f) {
  unsigned u = __float_as_uint(f);
  return (u & 0x80000000u) ? ~u : (u | 0x80000000u);
}
__device__ __forceinline__ float fdec(unsigned u) {
  unsigned b = (u & 0x80000000u) ? (u ^ 0x80000000u) : ~u;
  return __uint_as_float(b);
}

// One wave computes one 16x16 tile of Y = X @ W + bias (+ resid), D fixed = 128.
// Uses V_WMMA_F32_16X16X4_F32: A 16x4 (v2f/lane), B 4x16 (v2f/lane), C/D 16x16 (v8f/lane).
// Layout (ISA 7.12.2): VGPR v, lane-half h -> K = v + 2h; lane&15 = M (for A) / N (for B).
__global__ __launch_bounds__(128) void gemm_bias_wmma(
    const float* __restrict__ X, const float* __restrict__ W,
    const float* __restrict__ bias, const float* __restrict__ resid,
    float* __restrict__ Y, int M) {
  const int lane = threadIdx.x;       // 0..31 (wave32)
  const int half = lane >> 4;         // 0/1
  const int l15  = lane & 15;
  const int tileN = threadIdx.y + 4 * blockIdx.y;  // 0..7 -> covers 128 cols
  const int tileM = blockIdx.x;
  if (tileM * 16 >= M) return;        // wave-uniform guard (EXEC stays all-1 for WMMA)

  const float* Arow = X + (size_t)(tileM * 16 + l15) * DIM;
  const int col = tileN * 16 + l15;

  v8f c = {};
#pragma unroll
  for (int k = 0; k < DIM; k += 4) {
    v2f a = *(const v2f*)(Arow + k + 2 * half);           // A[m, k+2h .. k+2h+1]
    v2f b;
    b.x = W[(size_t)(k + 2 * half + 0) * DIM + col];      // B[k+2h+0, n]
    b.y = W[(size_t)(k + 2 * half + 1) * DIM + col];      // B[k+2h+1, n]
    c = __builtin_amdgcn_wmma_f32_16x16x4_f32(false, a, false, b, (short)0, c,
                                              false, false);
  }

  const float bv = bias[col];
#pragma unroll
  for (int r = 0; r < 8; ++r) {
    const int row = tileM * 16 + r + 8 * half;            // C/D: VGPR r -> M=r / r+8
    float o = c[r] + bv;
    if (resid) o += resid[(size_t)row * DIM + col];
    Y[(size_t)row * DIM + col] = o;
  }
}

__global__ void zero_u32(unsigned* __restrict__ p, size_t n) {
  size_t t = (size_t)blockIdx.x * blockDim.x + threadIdx.x;
  if (t < n) p[t] = 0u;
}

// One wave per edge: per-head dot(Q[src], K[dst]) via coalesced loads + shfl butterfly.
__global__ __launch_bounds__(256) void edge_scores(
    const float* __restrict__ Q, const float* __restrict__ Ke,
    const int* __restrict__ src, const int* __restrict__ dst,
    float* __restrict__ scores, unsigned* __restrict__ mbuf, int E) {
  const int wave = (int)((blockIdx.x * blockDim.x + threadIdx.x) >> 5);
  const int lane = threadIdx.x & 31;
  if (wave >= E) return;
  const int s = src[wave], d = dst[wave];
  const float* q = Q  + (size_t)s * DIM;
  const float* k = Ke + (size_t)d * DIM;

  float p[NH];
#pragma unroll
  for (int h = 0; h < NH; ++h)
    p[h] = q[h * HD + lane] * k[h * HD + lane];
#pragma unroll
  for (int h = 0; h < NH; ++h) {
#pragma unroll
    for (int off = 16; off > 0; off >>= 1)
      p[h] += __shfl_xor(p[h], off, 32);
  }
  if (lane == 0) {
    const float scale = 0.17677669529663687f;  // 1/sqrt(32)
#pragma unroll
    for (int h = 0; h < NH; ++h) {
      float sc = p[h] * scale;
      scores[(size_t)wave * NH + h] = sc;
      atomicMax(&mbuf[(size_t)s * NH + h], fenc(sc));
    }
  }
}

// Thread per (edge, head): e = exp(score - segmax); segment-sum into z.
__global__ void edge_exp(const int* __restrict__ src, float* __restrict__ scores,
                         const unsigned* __restrict__ mbuf, float* __restrict__ z,
                         int EH) {
  int t = blockIdx.x * blockDim.x + threadIdx.x;
  if (t >= EH) return;
  int e = t >> 2, h = t & 3;
  int s = src[e];
  float m = fdec(mbuf[(size_t)s * NH + h]);
  float ex = __expf(scores[t] - m);
  scores[t] = ex;
  atomicAdd(&z[(size_t)s * NH + h], ex);
}

// One wave per edge: msg = (e/z) * V[dst]; scatter-add into accum[src].
__global__ __launch_bounds__(256) void edge_scatter(
    const float* __restrict__ V, const int* __restrict__ src,
    const int* __restrict__ dst, const float* __restrict__ scores,
    const float* __restrict__ z, float* __restrict__ accum, int E) {
  const int wave = (int)((blockIdx.x * blockDim.x + threadIdx.x) >> 5);
  const int lane = threadIdx.x & 31;
  if (wave >= E) return;
  const int s = src[wave], d = dst[wave];
#pragma unroll
  for (int h = 0; h < NH; ++h) {
    float w = scores[(size_t)wave * NH + h] / z[(size_t)s * NH + h];
    float val = w * V[(size_t)d * DIM + h * HD + lane];
    atomicAdd(&accum[(size_t)s * DIM + h * HD + lane], val);
  }
}

extern "C" void kernel_launch(void* const* d_in, const int* in_sizes, int n_in,
                              void* d_out, int out_size, void* d_ws, size_t ws_size,
                              hipStream_t stream) {
  const float* x_node = (const float*)d_in[0];
  const float* x_edge = (const float*)d_in[1];
  const int*   src    = (const int*)d_in[2];
  const int*   dst    = (const int*)d_in[3];
  const float* Wq = (const float*)d_in[4];
  const float* bq = (const float*)d_in[5];
  const float* Wk = (const float*)d_in[6];
  const float* bk = (const float*)d_in[7];
  const float* Wv = (const float*)d_in[8];
  const float* bv = (const float*)d_in[9];
  const float* Wo = (const float*)d_in[10];
  const float* bo = (const float*)d_in[11];

  const int N  = in_sizes[0] / DIM;   // 50000 nodes
  const int NE = in_sizes[1] / DIM;   // 20000 hyperedges
  const int E  = in_sizes[2];         // 800000 edges

  // Workspace carve-out (all sizes multiples of 8B; m/z/accum contiguous for one zero pass)
  char* ws = (char*)d_ws;
  float* Q      = (float*)ws;  ws += (size_t)N  * DIM * sizeof(float);
  float* K      = (float*)ws;  ws += (size_t)NE * DIM * sizeof(float);
  float* V      = (float*)ws;  ws += (size_t)NE * DIM * sizeof(float);
  float* scores = (float*)ws;  ws += (size_t)E  * NH  * sizeof(float);
  unsigned* mbuf= (unsigned*)ws; ws += (size_t)N * NH * sizeof(unsigned);
  float* z      = (float*)ws;  ws += (size_t)N * NH * sizeof(float);
  float* accum  = (float*)ws;  ws += (size_t)N * DIM * sizeof(float);

  const dim3 gblk(32, 4);  // 4 waves/block, one 16x16 tile per wave

  // Projections (fp32 WMMA, bias fused)
  gemm_bias_wmma<<<dim3(N  / 16, 2), gblk, 0, stream>>>(x_node, Wq, bq, nullptr, Q, N);
  gemm_bias_wmma<<<dim3(NE / 16, 2), gblk, 0, stream>>>(x_edge, Wk, bk, nullptr, K, NE);
  gemm_bias_wmma<<<dim3(NE / 16, 2), gblk, 0, stream>>>(x_edge, Wv, bv, nullptr, V, NE);

  // Zero segmax encoding (0 == -inf under fenc), z, and accum in one contiguous sweep
  size_t nzero = (size_t)N * NH * 2 + (size_t)N * DIM;
  zero_u32<<<(unsigned)((nzero + 255) / 256), 256, 0, stream>>>(mbuf, nzero);

  // Edge phase (L2-resident intermediates: ~86 MB << 192 MB L2)
  const int wavesPerBlk = 8;  // 256 threads
  const int eblk = (E + wavesPerBlk - 1) / wavesPerBlk;
  edge_scores<<<eblk, 256, 0, stream>>>(Q, K, src, dst, scores, mbuf, E);
  const int EH = E * NH;
  edge_exp<<<(EH + 255) / 256, 256, 0, stream>>>(src, scores, mbuf, z, EH);
  edge_scatter<<<eblk, 256, 0, stream>>>(V, src, dst, scores, z, accum, E);

  // Output projection with fused bias + residual
  gemm_bias_wmma<<<dim3(N / 16, 2), gblk, 0, stream>>>(accum, Wo, bo, x_node,
                                                       (float*)d_out, N);
}